// SingleActionGNNPolicy_14199161881205
// MI455X (gfx1250) — compile-verified
//
#include <hip/hip_runtime.h>
#include <math.h>

typedef __attribute__((ext_vector_type(2))) float v2f;
typedef __attribute__((ext_vector_type(8))) float v8f;

#define D_DIM 128

// ---------------------------------------------------------------------------
// Kernel 1: logits[i] = dot(h[i,:], W) + b   via V_WMMA_F32_16X16X4_F32.
// One wave handles 16 rows. A = h tile (documented 16x4 f32 A layout:
// lanes 0-15 hold K=0,1 ; lanes 16-31 hold K=2,3 of the same 16 rows).
// B = W chunk broadcast along all 16 N columns -> every D column holds the
// identical 16-row dot product; lanes 0 and 16 write the results.
// W is staged in LDS so the hot loop issues exactly ONE global load (h
// fragment) per WMMA; W fragment reads ride the independent DS pipe.
// ---------------------------------------------------------------------------
__global__ __launch_bounds__(256) void gemv_logits_wmma(
    const float* __restrict__ h, const float* __restrict__ W,
    const float* __restrict__ bias, float* __restrict__ logits, int N)
{
  __shared__ float Wlds[D_DIM];
  if (threadIdx.x < D_DIM) Wlds[threadIdx.x] = W[threadIdx.x];
  __syncthreads();   // all waves participate (before any early exit)

  const int lane = threadIdx.x & 31;
  const int wave = (int)((blockIdx.x * blockDim.x + threadIdx.x) >> 5);
  const int tiles = (N + 15) >> 4;
  if (wave >= tiles) return;

  const int r0   = wave << 4;
  const int half = lane >> 4;   // 0: K=0,1   1: K=2,3
  const int m    = lane & 15;   // row within tile
  int row = r0 + m;
  if (row >= N) row = N - 1;    // clamp tail tile (keeps EXEC all-ones)
  const float* rp = h + (size_t)row * D_DIM + 2 * half;
  const float* wp = Wlds + 2 * half;

  v8f acc = {};
#pragma unroll
  for (int k0 = 0; k0 < D_DIM; k0 += 4) {
    v2f a = *(const v2f*)(rp + k0);   // h[row][k0+2*half .. +1]  (global)
    v2f bw;                           // B[k][n] = W[k0+k] broadcast in n (LDS)
    bw.x = wp[k0];
    bw.y = wp[k0 + 1];
    acc = __builtin_amdgcn_wmma_f32_16x16x4_f32(
        /*neg_a=*/false, a, /*neg_b=*/false, bw,
        /*c_mod=*/(short)0, acc, /*reuse_a=*/false, /*reuse_b=*/false);
  }

  // D layout: VGPR v, lanes 0-15 -> M=v ; lanes 16-31 -> M=8+v.
  if (m == 0) {
    const float bv = bias[0];
    const int base = r0 + 8 * half;
#pragma unroll
    for (int v = 0; v < 8; ++v) {
      const int r = base + v;
      if (r < N) logits[r] = acc[v] + bv;
    }
  }
}

// ---------------------------------------------------------------------------
// Kernel 2: ONE WAVE per graph (avg segment = N/G ~ 244 nodes; ~8/lane).
// batch_idx is sorted -> contiguous segment, found by binary search.
// Pass 1: segment max. Pass 2: sum z and sum z*(l-m), both reduced with
// wave32 shuffles (no LDS, no barriers).
// entropy = log(den) - (sum z*(l-m))/den
// ---------------------------------------------------------------------------
__global__ __launch_bounds__(256) void segment_softmax_stats(
    const float* __restrict__ logits, const long long* __restrict__ batch_idx,
    float* __restrict__ seg_max, float* __restrict__ seg_den,
    float* __restrict__ entropy_out, int N, int G)
{
  const int wave = (int)((blockIdx.x * blockDim.x + threadIdx.x) >> 5);
  const int lane = threadIdx.x & 31;
  if (wave >= G) return;
  const int g = wave;

  int lo = 0, hi = 0;
  if (lane == 0) {
    int a = 0, c = N;
    while (a < c) { int mid = (a + c) >> 1;
                    if (batch_idx[mid] < (long long)g) a = mid + 1; else c = mid; }
    lo = a; c = N;
    while (a < c) { int mid = (a + c) >> 1;
                    if (batch_idx[mid] < (long long)(g + 1)) a = mid + 1; else c = mid; }
    hi = a;
  }
  lo = __shfl(lo, 0, 32);
  hi = __shfl(hi, 0, 32);

  // ---- pass 1: max ----
  float mx = -INFINITY;
  for (int i = lo + lane; i < hi; i += 32) mx = fmaxf(mx, logits[i]);
#pragma unroll
  for (int off = 16; off > 0; off >>= 1)
    mx = fmaxf(mx, __shfl_xor(mx, off, 32));

  // ---- pass 2: sum z, sum z*(l-mx) ----
  float sz = 0.f, szl = 0.f;
  for (int i = lo + lane; i < hi; i += 32) {
    const float d = logits[i] - mx;
    const float z = expf(d);
    sz  += z;
    szl += z * d;
  }
#pragma unroll
  for (int off = 16; off > 0; off >>= 1) {
    sz  += __shfl_xor(sz,  off, 32);
    szl += __shfl_xor(szl, off, 32);
  }

  if (lane == 0) {
    seg_max[g] = mx;
    seg_den[g] = sz;
    entropy_out[g] = (hi > lo) ? (logf(sz) - szl / sz) : 0.f;
  }
}

// ---------------------------------------------------------------------------
// Kernel 3: logprob[g] = log p[actions[g]]
//         = logits[a] - seg_max[seg(a)] - log(den[seg(a)])
// (actions index GLOBAL nodes; must use the action's own segment.)
// ---------------------------------------------------------------------------
__global__ __launch_bounds__(256) void gather_logprob(
    const long long* __restrict__ actions, const long long* __restrict__ batch_idx,
    const float* __restrict__ logits, const float* __restrict__ seg_max,
    const float* __restrict__ seg_den, float* __restrict__ logprob, int G)
{
  const int g = blockIdx.x * blockDim.x + threadIdx.x;
  if (g >= G) return;
  const long long a = actions[g];
  const int s = (int)batch_idx[a];
  logprob[g] = logits[a] - seg_max[s] - logf(seg_den[s]);
}

// ---------------------------------------------------------------------------
extern "C" void kernel_launch(void* const* d_in, const int* in_sizes, int n_in,
                              void* d_out, int out_size, void* d_ws, size_t ws_size,
                              hipStream_t stream)
{
  const long long* actions   = (const long long*)d_in[0];
  const float*     h         = (const float*)d_in[1];
  const long long* batch_idx = (const long long*)d_in[2];
  const float*     W         = (const float*)d_in[3];
  const float*     b         = (const float*)d_in[4];

  const int N = in_sizes[2];   // num nodes
  const int G = in_sizes[0];   // num graphs

  float* out_logprob = (float*)d_out;       // [0, G)
  float* out_entropy = (float*)d_out + G;   // [G, 2G)

  float* ws_logits  = (float*)d_ws;         // N floats
  float* ws_seg_max = ws_logits + N;        // G floats
  float* ws_seg_den = ws_seg_max + G;       // G floats

  // Kernel 1: 1 wave per 16 rows, 8 waves per block.
  const int tiles  = (N + 15) / 16;
  const int blocks = (tiles + 7) / 8;
  gemv_logits_wmma<<<blocks, 256, 0, stream>>>(h, W, b, ws_logits, N);

  // Kernel 2: one wave per graph, 8 waves per block.
  const int blocks2 = (G + 7) / 8;
  segment_softmax_stats<<<blocks2, 256, 0, stream>>>(ws_logits, batch_idx,
                                                     ws_seg_max, ws_seg_den,
                                                     out_entropy, N, G);

  // Kernel 3: one thread per graph.
  gather_logprob<<<(G + 255) / 256, 256, 0, stream>>>(actions, batch_idx,
                                                      ws_logits, ws_seg_max,
                                                      ws_seg_den, out_logprob, G);
}